// SpatialNumDualDescriptorPM4_21440476741683
// MI455X (gfx1250) — compile-verified
//
#include <hip/hip_runtime.h>

typedef __attribute__((ext_vector_type(2))) float v2f;
typedef __attribute__((ext_vector_type(8))) float v8f;

#define M_DIM      32
#define O_DIM      19
#define NWIN       130321        // 19^4
#define CTAB_ELEMS (19 * 1024)   // C[k][i][j], k<19, (i,j) in 32x32
#define WIN_ELEMS  (NWIN * 32)

// ---------------------------------------------------------------------------
// Kernel 1: cosine table  C[k][i][j] = cos(2*pi*k / (i*32 + j + 2))
// Identical for all four axes (same extent 19, same periods).
// ---------------------------------------------------------------------------
__global__ void cos_table_kernel(float* __restrict__ Ctab) {
    int idx = blockIdx.x * blockDim.x + threadIdx.x;
    if (idx >= CTAB_ELEMS) return;
    int k = idx >> 10;          // 0..18
    int r = idx & 1023;         // i*32 + j
    float period = (float)(r + 2);
    const float TWO_PI = 6.28318530717958647692f;
    Ctab[idx] = cosf(TWO_PI * (float)k / period);
}

// ---------------------------------------------------------------------------
// Kernel 2: window average. win[w][0..31] = mean of 2^4 shifted 32-vectors.
// One thread per (window, 4-channel group): 16 float4 loads (hypervol is
// 20.5MB -> fully L2 resident, redundant reads are cheap).
// ---------------------------------------------------------------------------
__global__ void win_avg_kernel(const float* __restrict__ h, float* __restrict__ win) {
    int gid = blockIdx.x * blockDim.x + threadIdx.x;
    const int total = NWIN * 8;
    if (gid >= total) return;
    int w = gid >> 3;
    int q = gid & 7;                 // which float4 of the 32 channels
    int d = w % 19; int t = w / 19;
    int c = t % 19; t /= 19;
    int b = t % 19; int a = t / 19;
    // float4 strides: m=1(4 floats) -> s_d=8, s_c=160, s_b=3200, s_a=64000
    const float4* hv = (const float4*)h;
    int base = a * 64000 + b * 3200 + c * 160 + d * 8 + q;
    float sx = 0.f, sy = 0.f, sz = 0.f, sw = 0.f;
#pragma unroll
    for (int da = 0; da < 2; ++da)
#pragma unroll
        for (int db = 0; db < 2; ++db)
#pragma unroll
            for (int dc = 0; dc < 2; ++dc)
#pragma unroll
                for (int dd = 0; dd < 2; ++dd) {
                    float4 v = hv[base + da * 64000 + db * 3200 + dc * 160 + dd * 8];
                    sx += v.x; sy += v.y; sz += v.z; sw += v.w;
                }
    const float inv = 1.0f / 16.0f;
    float4 o; o.x = sx * inv; o.y = sy * inv; o.z = sz * inv; o.w = sw * inv;
    ((float4*)win)[w * 8 + q] = o;
}

// ---------------------------------------------------------------------------
// Kernel 3: x = win @ M_w^T  via V_WMMA_F32_16X16X4_F32 (f32, exact precision).
// One wave per 16-window tile; N=32 -> two 16x16 output tiles; K=32 -> 8 WMMAs
// per tile. A: row=lane%16 (window), K pair = 2*(lane/16)+{0,1}.
// B[k][n] = M_w[n][k] (row-major M_w), col=lane%16(+16), same K-pair scheme.
// D: M = v + 8*(lane/16), N = lane%16 (+16 for second tile).
// ---------------------------------------------------------------------------
__global__ void x_gemm_wmma_kernel(const float* __restrict__ win,
                                   const float* __restrict__ Mw,
                                   float* __restrict__ x,
                                   int nWin, int nTiles) {
    int wave = (blockIdx.x * blockDim.x + threadIdx.x) >> 5;
    if (wave >= nTiles) return;                 // uniform per wave
    int lane = threadIdx.x & 31;
    int row  = lane & 15;
    int hi   = lane >> 4;

    long w0 = (long)wave * 16;
    long wa = w0 + row;
    if (wa > (long)nWin - 1) wa = (long)nWin - 1;   // clamp: keep EXEC all-ones

    const float* arow = win + wa * 32 + 2 * hi;
    const float* b0p  = Mw + (long)row * 32 + 2 * hi;
    const float* b1p  = Mw + (long)(row + 16) * 32 + 2 * hi;

    v8f c0 = {0.f, 0.f, 0.f, 0.f, 0.f, 0.f, 0.f, 0.f};
    v8f c1 = {0.f, 0.f, 0.f, 0.f, 0.f, 0.f, 0.f, 0.f};

#pragma unroll
    for (int kk = 0; kk < 8; ++kk) {
        v2f af = *(const v2f*)(arow + kk * 4);
        v2f b0 = *(const v2f*)(b0p + kk * 4);
        v2f b1 = *(const v2f*)(b1p + kk * 4);
        c0 = __builtin_amdgcn_wmma_f32_16x16x4_f32(false, af, false, b0,
                                                   (short)0, c0, false, false);
        c1 = __builtin_amdgcn_wmma_f32_16x16x4_f32(false, af, false, b1,
                                                   (short)0, c1, false, false);
    }

#pragma unroll
    for (int v = 0; v < 8; ++v) {
        long wm = w0 + v + 8 * hi;
        if (wm < nWin) {
            x[wm * 32 + row]      = c0[v];
            x[wm * 32 + row + 16] = c1[v];
        }
    }
}

// ---------------------------------------------------------------------------
// Kernel 4: Nk[w,i] = sum_j P[i,j]*x[w,j]*C[a]*C[b]*C[c]*C[d]  (per i,j).
// Block = one (a,b,c): build G[i,j] = P*C[a]*C[b]*C[c] in LDS once, stage the
// 19 contiguous x rows in LDS, then each (warp=d, lane=i) does the 32-term
// j-reduction reading C[d] from global (hot in L0/L2).
// ---------------------------------------------------------------------------
__global__ void nk_kernel(const float* __restrict__ x,
                          const float* __restrict__ P,
                          const float* __restrict__ Ctab,
                          float* __restrict__ out) {
    int blk  = blockIdx.x;          // (a,b,c), c fastest
    int cidx = blk % 19; int t = blk / 19;
    int bidx = t % 19;   int aidx = t / 19;

    __shared__ float G[1024];
    __shared__ float xs[19 * 32];

    int tid = threadIdx.x;
    const float* Ca = Ctab + aidx * 1024;
    const float* Cb = Ctab + bidx * 1024;
    const float* Cc = Ctab + cidx * 1024;
    for (int idx = tid; idx < 1024; idx += 256)
        G[idx] = P[idx] * Ca[idx] * Cb[idx] * Cc[idx];

    long w0 = (((long)aidx * 19 + bidx) * 19 + cidx) * 19;  // first window (d=0)
    const float* xrow = x + w0 * 32;
    for (int idx = tid; idx < 19 * 32; idx += 256)
        xs[idx] = xrow[idx];

    __syncthreads();

    int lane = tid & 31;    // i
    int warp = tid >> 5;    // base d
    for (int d = warp; d < 19; d += 8) {
        const float* Cd = Ctab + d * 1024 + lane * 32;
        const float* Gi = G + lane * 32;
        const float* xd = xs + d * 32;
        float acc = 0.f;
#pragma unroll
        for (int j = 0; j < 32; ++j)
            acc += xd[j] * Gi[j] * Cd[j];
        out[(w0 + d) * 32 + lane] = acc;
    }
}

// ---------------------------------------------------------------------------
// Launch. ws layout (floats):
//   [0, 19456)                      cos table C
//   [19456, 19456+4170272)          win
//   [19456+4170272, +4170272)       x
// Total scratch ~33.5 MB.
// ---------------------------------------------------------------------------
extern "C" void kernel_launch(void* const* d_in, const int* in_sizes, int n_in,
                              void* d_out, int out_size, void* d_ws, size_t ws_size,
                              hipStream_t stream) {
    const float* hvol = (const float*)d_in[0];   // (20,20,20,20,32) f32
    const float* Mw   = (const float*)d_in[1];   // (32,32) f32
    const float* P    = (const float*)d_in[2];   // (32,32) f32
    float* out = (float*)d_out;                  // (130321, 32) f32

    float* wsf  = (float*)d_ws;
    float* Ctab = wsf;
    float* win  = wsf + CTAB_ELEMS;
    float* xbuf = wsf + CTAB_ELEMS + WIN_ELEMS;

    (void)in_sizes; (void)n_in; (void)out_size; (void)ws_size;

    // 1) cosine table: 19456 elems
    cos_table_kernel<<<(CTAB_ELEMS + 255) / 256, 256, 0, stream>>>(Ctab);

    // 2) window average: 130321 windows * 8 float4-groups
    win_avg_kernel<<<(NWIN * 8 + 255) / 256, 256, 0, stream>>>(hvol, win);

    // 3) WMMA GEMM: ceil(130321/16) = 8146 tiles, 8 waves/block
    int nTiles = (NWIN + 15) / 16;
    int gBlocks = (nTiles + 7) / 8;
    x_gemm_wmma_kernel<<<gBlocks, 256, 0, stream>>>(win, Mw, xbuf, NWIN, nTiles);

    // 4) final reduction: one block per (a,b,c) = 19^3 blocks
    nk_kernel<<<19 * 19 * 19, 256, 0, stream>>>(xbuf, P, Ctab, out);
}